// NLinear_9268539425052
// MI455X (gfx1250) — compile-verified
//
#include <hip/hip_runtime.h>

// Problem constants from the reference
#define B_ 1024
#define S_ 336
#define C_ 321
#define P_ 96

typedef __attribute__((ext_vector_type(2))) float v2f;
typedef __attribute__((ext_vector_type(8))) float v8f;

// ---------------------------------------------------------------------------
// Phase 1: gather channel 3 of x, subtract the per-row last element, and
// compact into workspace.  xn[b*S + s] = x[b,s,3] - x[b,S-1,3];
// lastArr[b] = x[b,S-1,3].  One thread per (b,s) -> 10752 waves of pure
// gather, the best we can do for a stride-321-float access pattern.
// ---------------------------------------------------------------------------
__global__ void nlinear_extract(const float* __restrict__ x,
                                float* __restrict__ xn,
                                float* __restrict__ lastArr) {
    int tid = blockIdx.x * blockDim.x + threadIdx.x;
    if (tid >= B_ * S_) return;
    int b = tid / S_;
    int s = tid - b * S_;
    float lastv = x[((size_t)b * S_ + (S_ - 1)) * C_ + 3];  // broadcast within row
    float v     = x[((size_t)b * S_ + s) * C_ + 3];
    xn[tid] = v - lastv;
    if (s == S_ - 1) lastArr[b] = lastv;
}

// ---------------------------------------------------------------------------
// Phase 2: out(1024x96) = xn(1024x336) * W3^T(336x96) + bias + last,
// using V_WMMA_F32_16X16X4_F32.  One wave per 16x16 output tile:
// 64 M-tiles x 6 N-tiles = 384 waves, K-loop of 84 WMMA steps.
//
// A 16x4 f32 layout: lane holds row (lane&15); VGPR0/1 = K pair starting at
// k + 2*(lane>>4).  B 4x16: lane holds col (lane&15), same K pair.  Both are
// aligned float2 loads.  C/D: VGPR r -> row r + 8*(lane>>4), col lane&15.
//
// COMPACT=false is a fused fallback (reads x directly, strided) used only if
// ws_size is too small for the compact xn buffer.
// ---------------------------------------------------------------------------
template <bool COMPACT>
__global__ void nlinear_gemm(const float* __restrict__ A,      // xn (compact) or x (raw)
                             const float* __restrict__ Wfull,  // (C,P,S)
                             const float* __restrict__ bias,   // (C,P)
                             const float* __restrict__ lastArr,// last (compact) or unused
                             float* __restrict__ out) {        // (B,P)
    const int lane = threadIdx.x & 31;
    const int wave = blockIdx.x * (blockDim.x >> 5) + (threadIdx.x >> 5);
    const int mt   = wave / 6;
    const int nt   = wave - mt * 6;
    const int half = lane >> 4;
    const int lm   = lane & 15;

    const int m = mt * 16 + lm;   // A row for this lane
    const int n = nt * 16 + lm;   // B col (and C/D col) for this lane

    // W[3][n][s] is contiguous in s
    const float* __restrict__ wrow = Wfull + ((size_t)(3 * P_) + n) * S_;

    const float* __restrict__ arow;
    float lastA = 0.0f;
    if (COMPACT) {
        arow = A + (size_t)m * S_;
    } else {
        arow  = A + ((size_t)m * S_) * C_ + 3;
        lastA = A[((size_t)m * S_ + (S_ - 1)) * C_ + 3];
    }

    v8f acc = {};
#pragma unroll 4
    for (int k = 0; k < S_; k += 4) {
        const int k0 = k + 2 * half;     // this lane's K pair start
        v2f a, b;
        if (COMPACT) {
            a = *(const v2f*)(arow + k0);            // 8B-aligned (S even, k0 even)
        } else {
            a.x = arow[(size_t)k0 * C_] - lastA;     // strided fallback
            a.y = arow[(size_t)(k0 + 1) * C_] - lastA;
        }
        b = *(const v2f*)(wrow + k0);                // 8B-aligned
        // D = A(16x4) * B(4x16) + C, f32 throughout
        acc = __builtin_amdgcn_wmma_f32_16x16x4_f32(
            /*neg_a=*/false, a, /*neg_b=*/false, b,
            /*c_mod=*/(short)0, acc, /*reuse_a=*/false, /*reuse_b=*/false);
    }

    const float bv = bias[3 * P_ + n];   // b[3, n]
#pragma unroll
    for (int r = 0; r < 8; ++r) {
        const int row = mt * 16 + r + 8 * half;      // C/D row for VGPR r
        float lastR;
        if (COMPACT) lastR = lastArr[row];
        else         lastR = A[((size_t)row * S_ + (S_ - 1)) * C_ + 3];
        out[(size_t)row * P_ + n] = acc[r] + bv + lastR;
    }
}

extern "C" void kernel_launch(void* const* d_in, const int* in_sizes, int n_in,
                              void* d_out, int out_size, void* d_ws, size_t ws_size,
                              hipStream_t stream) {
    const float* x    = (const float*)d_in[0];   // (B,S,C) f32
    const float* W    = (const float*)d_in[1];   // (C,P,S) f32
    const float* bias = (const float*)d_in[2];   // (C,P)   f32
    float*       out  = (float*)d_out;           // (B,P)   f32

    const size_t needed = (size_t)(B_ * S_ + B_) * sizeof(float);

    // 384 waves, 4 per block of 128 threads -> 96 blocks
    const int gemm_blocks  = (64 * 6) / 4;
    const int gemm_threads = 128;

    if (ws_size >= needed) {
        float* xn      = (float*)d_ws;
        float* lastArr = xn + (size_t)B_ * S_;
        const int n_ext = B_ * S_;
        nlinear_extract<<<(n_ext + 255) / 256, 256, 0, stream>>>(x, xn, lastArr);
        nlinear_gemm<true><<<gemm_blocks, gemm_threads, 0, stream>>>(
            xn, W, bias, lastArr, out);
    } else {
        nlinear_gemm<false><<<gemm_blocks, gemm_threads, 0, stream>>>(
            x, W, bias, nullptr, out);
    }
}